// RadialFunction_52080773431864
// MI455X (gfx1250) — compile-verified
//
#include <hip/hip_runtime.h>
#include <math.h>

#define N_SPECIES 119
#define N_PAIRS   (119*119)      /* 14161 */
#define N_BASIS   16
#define N_RADIAL  8

typedef float v2f __attribute__((ext_vector_type(2)));
typedef float v8f __attribute__((ext_vector_type(8)));

// ---------------- binning: histogram -> scan -> scatter ----------------

__global__ void hist_kernel(const int* __restrict__ Zi, const int* __restrict__ Zj,
                            unsigned* __restrict__ counts, int n) {
  int e = blockIdx.x * blockDim.x + threadIdx.x;
  if (e < n) {
    int pair = Zj[e] * N_SPECIES + Zi[e];   // embeddings[Z_j, Z_i] order
    atomicAdd(&counts[pair], 1u);
  }
}

__global__ void scan_kernel(const unsigned* __restrict__ counts,
                            unsigned* __restrict__ offsets,
                            unsigned* __restrict__ cursor) {
  __shared__ unsigned csum[256];
  int t = threadIdx.x;
  const int per = (N_PAIRS + 255) / 256;
  int lo = t * per;
  int hi = lo + per; if (hi > N_PAIRS) hi = N_PAIRS;
  unsigned s = 0u;
  for (int i = lo; i < hi; ++i) s += counts[i];
  csum[t] = s;
  __syncthreads();
  if (t == 0) {
    unsigned run = 0u;
    for (int i = 0; i < 256; ++i) { unsigned c = csum[i]; csum[i] = run; run += c; }
  }
  __syncthreads();
  unsigned run = csum[t];
  for (int i = lo; i < hi; ++i) { offsets[i] = run; cursor[i] = run; run += counts[i]; }
}

__global__ void scatter_kernel(const int* __restrict__ Zi, const int* __restrict__ Zj,
                               unsigned* __restrict__ cursor, unsigned* __restrict__ sorted,
                               int n) {
  int e = blockIdx.x * blockDim.x + threadIdx.x;
  if (e < n) {
    int pair = Zj[e] * N_SPECIES + Zi[e];
    unsigned pos = atomicAdd(&cursor[pair], 1u);
    sorted[pos] = (unsigned)e;
  }
}

// ---------------- per-pair batched GEMM via V_WMMA_F32_16X16X4_F32 ----------------
//
// D[16x16] += A[16x4] * B[4x16], chained 4x over K to cover the 16 basis fns.
//   A[m][k]  = rad_norm * cutoff[e_m] * exp(-betta*(shift[k]-dr[e_m])^2)   (m = edge in tile)
//   B[k][n]  = embed_norm * E[pair][n][k]  for n<8, else 0                 (n = radial idx)
// ISA 7.12.2 layouts (wave32):
//   A (16x4, 2 VGPR):  M = lane&15,  K = v + 2*(lane>>4)
//   B (4x16, 2 VGPR):  N = lane&15,  K = v + 2*(lane>>4)
//   D (16x16, 8 VGPR): N = lane&15,  M = v + 8*(lane>>4)

__global__ __launch_bounds__(256) void wmma_radial_kernel(
    const float* __restrict__ dr, const float* __restrict__ cutoff,
    const float* __restrict__ emb,
    const unsigned* __restrict__ offsets, const unsigned* __restrict__ counts,
    const unsigned* __restrict__ sorted,
    float* __restrict__ out, float betta, float rad_norm) {
  const int pair = blockIdx.x;
  const unsigned beg = offsets[pair];
  const unsigned cnt = counts[pair];
  if (cnt == 0u) return;
  const unsigned end = beg + cnt;

  const int lane = threadIdx.x & 31;
  const int wave = threadIdx.x >> 5;
  const int kh   = lane >> 4;      // 0 or 1: selects K pair {2kh, 2kh+1} of each 4-K step
  const int ln   = lane & 15;      // M for A-frag, N for B/D-frags

  // B fragments: one pair's 8x16 coeff block, loaded once, held in 8 VGPRs.
  const float* E = emb + (size_t)pair * (N_RADIAL * N_BASIS);
  v2f bf[4];
#pragma unroll
  for (int kk = 0; kk < 4; ++kk) {
    const int k0 = kk * 4 + kh * 2;
    float bx = 0.f, by = 0.f;
    if (ln < N_RADIAL) {
      bx = 0.25f * E[ln * N_BASIS + k0];       // embed_norm = 1/sqrt(16) = 0.25 exactly
      by = 0.25f * E[ln * N_BASIS + k0 + 1];
    }
    bf[kk].x = bx; bf[kk].y = by;
  }

  const unsigned ntiles = (cnt + 15u) >> 4;
  for (unsigned tile = (unsigned)wave; tile < ntiles; tile += 8u) {
    const unsigned base = beg + tile * 16u;
    const unsigned epos = base + (unsigned)ln;
    unsigned eid = 0u; float mydr = 0.f, sca = 0.f;
    if (epos < end) {
      eid  = sorted[epos];
      mydr = dr[eid];
      sca  = rad_norm * cutoff[eid];           // fold cutoff & rad_norm into A rows
    }                                          // tail edges: sca=0 -> zero A rows

    v8f c = {0.f, 0.f, 0.f, 0.f, 0.f, 0.f, 0.f, 0.f};
#pragma unroll
    for (int kk = 0; kk < 4; ++kk) {
      const int k0 = kk * 4 + kh * 2;
      const float sh0 = 0.5f + 0.34375f * (float)k0;   // shift[b] = 0.5 + (5.5/16)*b, exact dyadic
      const float sh1 = sh0 + 0.34375f;
      const float d0 = sh0 - mydr;
      const float d1 = sh1 - mydr;
      v2f a;
      a.x = sca * __expf(-betta * d0 * d0);
      a.y = sca * __expf(-betta * d1 * d1);
      c = __builtin_amdgcn_wmma_f32_16x16x4_f32(false, a, false, bf[kk],
                                                (short)0, c, false, false);
    }

    // Scatter D: lanes with N=ln<8 hold radial r=ln for edge M = v + 8*kh.
#pragma unroll
    for (int v = 0; v < 8; ++v) {
      const int m = v + kh * 8;
      const unsigned ep = base + (unsigned)m;
      const unsigned oid = (unsigned)__shfl((int)eid, m, 32);  // edge id held by lane m
      if (ln < N_RADIAL && ep < end) {
        out[(size_t)oid * N_RADIAL + ln] = c[v];
      }
    }
  }
}

// ---------------- fallback (workspace too small): plain VALU path ----------------

__global__ void fallback_kernel(const float* __restrict__ dr, const int* __restrict__ Zi,
                                const int* __restrict__ Zj, const float* __restrict__ cutoff,
                                const float* __restrict__ emb, float* __restrict__ out,
                                int n, float betta, float rad_norm) {
  int e = blockIdx.x * blockDim.x + threadIdx.x;
  if (e >= n) return;
  float basis[N_BASIS];
  const float r = dr[e];
#pragma unroll
  for (int b = 0; b < N_BASIS; ++b) {
    float d = (0.5f + 0.34375f * (float)b) - r;
    basis[b] = rad_norm * __expf(-betta * d * d);
  }
  const float* E = emb + (size_t)(Zj[e] * N_SPECIES + Zi[e]) * (N_RADIAL * N_BASIS);
  const float cut = 0.25f * cutoff[e];
#pragma unroll
  for (int rr = 0; rr < N_RADIAL; ++rr) {
    float acc = 0.f;
#pragma unroll
    for (int b = 0; b < N_BASIS; ++b) acc += E[rr * N_BASIS + b] * basis[b];
    out[(size_t)e * N_RADIAL + rr] = acc * cut;
  }
}

// ---------------- launcher ----------------

extern "C" void kernel_launch(void* const* d_in, const int* in_sizes, int n_in,
                              void* d_out, int out_size, void* d_ws, size_t ws_size,
                              hipStream_t stream) {
  const float* dr     = (const float*)d_in[0];
  const int*   Zi     = (const int*)  d_in[1];
  const int*   Zj     = (const int*)  d_in[2];
  const float* cutoff = (const float*)d_in[3];
  const float* emb    = (const float*)d_in[4];
  float* out = (float*)d_out;
  const int n = in_sizes[0];

  const double betta_d = 256.0 / 36.0;                   // N_BASIS^2 / R_MAX^2
  const float  betta    = (float)betta_d;
  const float  rad_norm = (float)pow(2.0 * betta_d / M_PI, 0.25);

  const int blocks = (n + 255) / 256;
  const size_t need = ((size_t)3 * N_PAIRS + (size_t)n) * sizeof(unsigned);
  if (ws_size < need) {
    fallback_kernel<<<blocks, 256, 0, stream>>>(dr, Zi, Zj, cutoff, emb, out, n,
                                                betta, rad_norm);
    return;
  }

  unsigned* counts  = (unsigned*)d_ws;
  unsigned* offsets = counts  + N_PAIRS;
  unsigned* cursor  = offsets + N_PAIRS;
  unsigned* sorted  = cursor  + N_PAIRS;

  hipMemsetAsync(counts, 0, (size_t)N_PAIRS * sizeof(unsigned), stream);
  hist_kernel<<<blocks, 256, 0, stream>>>(Zi, Zj, counts, n);
  scan_kernel<<<1, 256, 0, stream>>>(counts, offsets, cursor);
  scatter_kernel<<<blocks, 256, 0, stream>>>(Zi, Zj, cursor, sorted, n);
  wmma_radial_kernel<<<N_PAIRS, 256, 0, stream>>>(dr, cutoff, emb, offsets, counts,
                                                  sorted, out, betta, rad_norm);
}